// Sobelxy_hd_5746666242606
// MI455X (gfx1250) — compile-verified
//
#include <hip/hip_runtime.h>

typedef float v4f __attribute__((ext_vector_type(4)));

#define TILE_H   32           // output rows per block (256/32 = 8 exact tiles)
#define HALO_H   34           // TILE_H + 2 halo rows staged in LDS
#define IMG_H    256
#define IMG_W    256
#define STRIDE   268          // LDS row stride (floats); image col c at idx c+4,
                              // zeros at idx 3 and 260 -> no edge guards in compute
#define NTHREADS 256          // 8 wave32s
#define NCHUNKS  (HALO_H * (IMG_W / 4) / 4 * 4) // 34 rows * 64 x 16B chunks = 2176

__global__ __launch_bounds__(NTHREADS)
void sobelxy_kernel(const float* __restrict__ x, float* __restrict__ out) {
    __shared__ float tile[HALO_H * STRIDE];   // ~36.4 KB

    const int p  = blockIdx.y;              // (b*c) plane
    const int y0 = blockIdx.x * TILE_H;     // first output row of tile
    const int t  = threadIdx.x;

    const size_t plane = (size_t)p * (size_t)(IMG_H * IMG_W);
    const float* xp = x + plane;
    float*       op = out + plane;

    // ---- 1. zero border columns (image col -1 and IMG_W) of every LDS row ----
    if (t < 2 * HALO_H) {
        const int r = t >> 1;
        const int c = (t & 1) ? (IMG_W + 4) : 3;
        tile[r * STRIDE + c] = 0.0f;
    }

    // ---- 2. async global -> LDS copy of 34 halo rows (CDNA5 async pipe) ----
    // 3 x 256 = 768 16B chunks over 256 threads; excess lanes clamp to the last chunk so
    // every async issue is full-EXEC (identical duplicate copies are benign).
    #pragma unroll
    for (int i = 0; i < 3; ++i) {
        int chunk = t + i * NTHREADS;
        chunk = chunk < (NCHUNKS - 1) ? chunk : (NCHUNKS - 1);
        const int r   = chunk >> 4;          // LDS row 0..47
        const int c16 = chunk & 15;          // 16B chunk within row
        int gy = y0 - 1 + r;                 // clamp vertical OOB; fixed in step 3
        gy = gy < 0 ? 0 : (gy > IMG_H - 1 ? IMG_H - 1 : gy);
        const float* gsrc = xp + (size_t)gy * IMG_W + c16 * 4;
        const unsigned ldsaddr =
            (unsigned)(uintptr_t)(&tile[r * STRIDE + 4 + c16 * 4]); // 16B aligned
        asm volatile("global_load_async_to_lds_b128 %0, %1, off"
                     :
                     : "v"(ldsaddr), "v"(gsrc)
                     : "memory");
    }
#if __has_builtin(__builtin_amdgcn_s_wait_asynccnt)
    __builtin_amdgcn_s_wait_asynccnt(0);
#else
    asm volatile("s_wait_asynccnt 0" ::: "memory");
#endif
    __syncthreads();

    // ---- 3. zero the (at most one) out-of-image halo row actually read ----
    if (y0 == 0)              tile[0 * STRIDE + 4 + t] = 0.0f;              // gy == -1
    if (y0 == IMG_H - TILE_H) tile[(IMG_H + 1 - y0) * STRIDE + 4 + t] = 0.0f; // gy == 256
    __syncthreads();

    // ---- 4. compute: each lane does two 4x4 output strips ----
    // Horizontal separable pass (hs = smooth, hd = derivative) is computed once
    // per LDS row and reused by the 3 output rows that touch it: ~1.1 ds-loads
    // and ~11 VALU ops per pixel, zero branches, zero cndmasks.
    #pragma unroll
    for (int i = 0; i < 2; ++i) {
        const int strip = t + i * NTHREADS;   // 0..511 (8 row-strips x 64 col-chunks)
        const int rs = strip >> 6;            // strip row 0..7
        const int c0 = (strip & 63) * 4;      // first output column

        float hs[6][4], hd[6][4];
        #pragma unroll
        for (int rr = 0; rr < 6; ++rr) {
            const float* row = &tile[(4 * rs + rr) * STRIDE];
            const v4f m = *(const v4f*)&row[c0 + 4];      // cols c0..c0+3 (16B aligned)
            const float v0 = row[c0 + 3];                 // col c0-1 (zero at c0==0)
            const float v5 = row[c0 + 8];                 // col c0+4 (zero at c0==252)
            const float v[6] = { v0, m.x, m.y, m.z, m.w, v5 };
            #pragma unroll
            for (int k = 0; k < 4; ++k) {
                hs[rr][k] = 3.0f * v[k] + 10.0f * v[k + 1] + 3.0f * v[k + 2];
                hd[rr][k] = v[k] - v[k + 2];
            }
        }

        #pragma unroll
        for (int k = 0; k < 4; ++k) {         // 4 output rows of the strip
            v4f o;
            #pragma unroll
            for (int j = 0; j < 4; ++j) {
                const float gx = 3.0f * hd[k][j] + 10.0f * hd[k + 1][j] + 3.0f * hd[k + 2][j];
                const float gy = hs[k][j] - hs[k + 2][j];
                o[j] = __builtin_fabsf(gx) + __builtin_fabsf(gy);  // folds to |a|+|b| add
            }
            // streamed output: non-temporal 128-bit store keeps the 192MB L2 clean
            v4f* dst = (v4f*)(op + (size_t)(y0 + 4 * rs + k) * IMG_W + c0);
            __builtin_nontemporal_store(o, dst);
        }
    }
}

extern "C" void kernel_launch(void* const* d_in, const int* in_sizes, int n_in,
                              void* d_out, int out_size, void* d_ws, size_t ws_size,
                              hipStream_t stream) {
    const float* x   = (const float*)d_in[0];
    float*       out = (float*)d_out;

    const int planes  = in_sizes[0] / (IMG_H * IMG_W);  // b*c = 1024
    const int tiles_y = IMG_H / TILE_H;                 // 8 (exact)

    dim3 grid(tiles_y, planes);
    dim3 block(NTHREADS);
    sobelxy_kernel<<<grid, block, 0, stream>>>(x, out);
}